// Warper3d_50405736185920
// MI455X (gfx1250) — compile-verified
//
#include <hip/hip_runtime.h>
#include <hip/hip_bf16.h>

// Warper3d: 3D trilinear grid-sample with border padding.
// img:  [B=4, 1, D=64, H=192, W=192] float32
// flow: [B, 3, D, H, W] float32
// out:  [B, 1, D, H, W] float32
//
// Memory-bound gather (~190 MB HBM -> ~8us floor @ 23.3 TB/s); VALU budget
// now well under that. This revision:
//  - char* + u32 byte offsets for ALL accesses (flow/out included) so every
//    load/store folds to saddr+voffset GVS form (no per-lane 64-bit math)
//  - coordinate math pre-folded: per element 1 FMA per axis
//  - x-neighbor pairs as one unaligned b64 gather (4 gathers/element);
//    border handled by -4B shifted load + weight override (fract==0 there)
//  - NT hints on stream-once flow loads / out stores; img gathers RT-cached

typedef __attribute__((ext_vector_type(4))) float v4f;
typedef __attribute__((ext_vector_type(2))) float v2f;

__device__ __forceinline__ v2f ld2(const char* __restrict__ base, unsigned off) {
  v2f r;
  __builtin_memcpy(&r, base + off, 8);   // unaligned <2 x float> global load
  return r;
}

__global__ __launch_bounds__(256) void warp3d_kernel(
    const float* __restrict__ img,
    const float* __restrict__ flow,
    float* __restrict__ out) {
  constexpr int Dn = 64, Hn = 192, Wn = 192;
  constexpr int HW  = Hn * Wn;        // 36864
  constexpr int DHW = Dn * HW;        // 2359296
  constexpr int VEC = 4;
  constexpr int NVEC_ROW = Wn / VEC;  // 48

  const float SX = (float)Wn / (float)(Wn - 1);
  const float SY = (float)Hn / (float)(Hn - 1);
  const float SZ = (float)Dn / (float)(Dn - 1);

  // Uniform (SALU): blockIdx.y = b*Dn + z
  int zb = blockIdx.y;
  int z  = zb & (Dn - 1);
  int b  = zb >> 6;

  // Per-thread: one z-plane = 9216 vec4 items = 36 blocks of 256
  int t  = blockIdx.x * 256 + threadIdx.x;   // 0..9215
  int y  = t / NVEC_ROW;
  int wv = t - y * NVEC_ROW;
  int w0 = wv * VEC;

  unsigned syB = (unsigned)(y * Wn + w0) << 2;   // byte offset in one z-plane

  // Uniform SGPR bases, all as char* so every access is base + zext(u32)
  const char* fb0 = (const char*)(flow + (b * 3) * DHW + z * HW);
  const char* fb1 = fb0 + (size_t)DHW * 4;
  const char* fb2 = fb1 + (size_t)DHW * 4;
  const char* ibc = (const char*)(img + b * DHW);
  char*       obc = (char*)(out + b * DHW + z * HW);

  v4f fx = __builtin_nontemporal_load((const v4f*)(fb0 + syB));
  v4f fy = __builtin_nontemporal_load((const v4f*)(fb1 + syB));
  v4f fz = __builtin_nontemporal_load((const v4f*)(fb2 + syB));

  // Pre-folded coordinate constants: (c + f)*S - 0.5 == f*S + (c*S - 0.5)
  float xc0 = fmaf((float)w0, SX, -0.5f);
  float xck[VEC] = {xc0, xc0 + SX, xc0 + 2.0f * SX, xc0 + 3.0f * SX};
  float yc = fmaf((float)y, SY, -0.5f);
  float zc = fmaf((float)z, SZ, -0.5f);   // uniform

  float fxa[VEC] = {fx.x, fx.y, fx.z, fx.w};
  float fya[VEC] = {fy.x, fy.y, fy.z, fy.w};
  float fza[VEC] = {fz.x, fz.y, fz.z, fz.w};
  float res[VEC];

#pragma unroll
  for (int k = 0; k < VEC; ++k) {
    float ix = __builtin_amdgcn_fmed3f(fmaf(fxa[k], SX, xck[k]),
                                       0.0f, (float)(Wn - 1));
    float iy = __builtin_amdgcn_fmed3f(fmaf(fya[k], SY, yc),
                                       0.0f, (float)(Hn - 1));
    float iz = __builtin_amdgcn_fmed3f(fmaf(fza[k], SZ, zc),
                                       0.0f, (float)(Dn - 1));

    // coords >= 0 -> trunc == floor; v_fract gives lerp weights
    int x0 = (int)ix, y0 = (int)iy, z0 = (int)iz;
    float wx = __builtin_amdgcn_fractf(ix);
    float wy = __builtin_amdgcn_fractf(iy);
    float wz = __builtin_amdgcn_fractf(iz);

    bool bx = ix < (float)(Wn - 1);
    bool by = iy < (float)(Hn - 1);
    bool bz = iz < (float)(Dn - 1);

    // Byte offsets; uniform base + zext(u32) -> saddr-form global loads.
    unsigned baseB = ((unsigned)((z0 * Hn + y0) * Wn + x0)) << 2;
    unsigned o   = baseB + (bx ? 0u : (unsigned)-4);  // shift back at border
    unsigned dyB = by ? (unsigned)(Wn * 4) : 0u;
    unsigned dzB = bz ? (unsigned)(HW * 4) : 0u;

    v2f P00 = ld2(ibc, o);              // {v000, v001} (or {prev, v000})
    v2f P01 = ld2(ibc, o + dyB);        // {v010, v011}
    v2f P10 = ld2(ibc, o + dzB);        // {v100, v101}
    v2f P11 = ld2(ibc, o + dyB + dzB);  // {v110, v111}

    // Interior: u = wx, pair = {v000,v001} -> standard lerp.
    // Border (bx=0, wx==0): u = 1, pair = {prev, v000} -> picks hi = v000.
    float u = bx ? wx : 1.0f;

    float c00 = fmaf(u, P00.y - P00.x, P00.x);
    float c01 = fmaf(u, P01.y - P01.x, P01.x);
    float c10 = fmaf(u, P10.y - P10.x, P10.x);
    float c11 = fmaf(u, P11.y - P11.x, P11.x);
    float c0  = fmaf(wy, c01 - c00, c00);
    float c1  = fmaf(wy, c11 - c10, c10);
    res[k]    = fmaf(wz, c1 - c0, c0);
  }

  v4f o4 = {res[0], res[1], res[2], res[3]};
  __builtin_nontemporal_store(o4, (v4f*)(obc + syB));
}

extern "C" void kernel_launch(void* const* d_in, const int* in_sizes, int n_in,
                              void* d_out, int out_size, void* d_ws, size_t ws_size,
                              hipStream_t stream) {
  const float* img  = (const float*)d_in[0];
  const float* flow = (const float*)d_in[1];
  float* out = (float*)d_out;

  dim3 grid(36, 4 * 64);   // (plane blocks, B*D)
  dim3 block(256);         // 8 wave32s

  warp3d_kernel<<<grid, block, 0, stream>>>(img, flow, out);
}